// CRF_38405597560860
// MI455X (gfx1250) — compile-verified
//
#include <hip/hip_runtime.h>
#include <hip/hip_bf16.h>

typedef __attribute__((ext_vector_type(2))) float v2f;
typedef __attribute__((ext_vector_type(8))) float v8f;

#define KTAG   48
#define START  46
#define STOP   47
#define NEGVAL -10000.0f
#define FVS    52        // padded LDS row stride (floats): 52*4B, 16 rows hit 16 distinct banks

// ---------------------------------------------------------------------------
// Kernel 0: Pt[i][j] = exp(trans[j][i])  (transposed so B-fragment rows are K=i)
// ---------------------------------------------------------------------------
__global__ void crf_prep(const float* __restrict__ trans, float* __restrict__ Pt) {
    int idx = blockIdx.x * blockDim.x + threadIdx.x;
    if (idx < KTAG * KTAG) {
        int i = idx / KTAG, j = idx % KTAG;
        Pt[i * KTAG + j] = __expf(trans[j * KTAG + i]);   // exp(-10000) underflows to 0
    }
}

// ---------------------------------------------------------------------------
// Kernel 1: forward scan. One wave32 per 16 batches. Per step:
//   rescale -> exp -> (16x48)x(48x48) GEMM via v_wmma_f32_16x16x4_f32 -> log
// ---------------------------------------------------------------------------
__global__ __launch_bounds__(32) void crf_scan(const float* __restrict__ feats,
                                               const int*   __restrict__ seq_len,
                                               const float* __restrict__ trans,
                                               const float* __restrict__ Pt,
                                               float*       __restrict__ fwd,
                                               int T) {
    const int lane = threadIdx.x;       // 0..31, single wave
    const int half = lane >> 4;         // 0 | 1
    const int l15  = lane & 15;
    const int bb   = blockIdx.x * 16;   // batch tile base

    __shared__ float fvLDS[16 * FVS];
    __shared__ float eLDS [16 * FVS];
    __shared__ float mpart[32];
    __shared__ float mLDS [16];
    __shared__ float tS   [KTAG];       // trans[STOP, j]

    // Constant B fragments: B[k=i][n=j] = Pt[i*48 + j]; 4x16 f32 layout:
    // VGPR0 = rows k0|k0+2 across lane halves, VGPR1 = rows k0+1|k0+3.
    v2f Bf[12][3];
#pragma unroll
    for (int c = 0; c < 12; ++c)
#pragma unroll
        for (int jt = 0; jt < 3; ++jt) {
            int k0 = 4 * c + 2 * half;
            int j  = jt * 16 + l15;
            Bf[c][jt].x = Pt[(k0 + 0) * KTAG + j];
            Bf[c][jt].y = Pt[(k0 + 1) * KTAG + j];
        }

    // init fv = all_fv[0]
#pragma unroll
    for (int q = 0; q < 24; ++q) {
        int j = half * 24 + q;
        fvLDS[l15 * FVS + j] = (j == START) ? 0.0f : NEGVAL;
    }
    for (int j = lane; j < KTAG; j += 32) tS[j] = trans[STOP * KTAG + j];
    __syncthreads();

    for (int t = 0; t < T; ++t) {
        // Prefetch emit for this step in C-fragment order (M = v + 8*half, N = l15).
        float emit[3][8];
#pragma unroll
        for (int jt = 0; jt < 3; ++jt)
#pragma unroll
            for (int v = 0; v < 8; ++v) {
                int M = v + 8 * half;
                emit[jt][v] = feats[(size_t)(bb + M) * T * KTAG + (size_t)t * KTAG + jt * 16 + l15];
            }

        // Row max: lane handles row l15, columns [half*24, half*24+24)
        float mp = -3.402823466e38f;
#pragma unroll
        for (int q = 0; q < 24; ++q)
            mp = fmaxf(mp, fvLDS[l15 * FVS + half * 24 + q]);
        mpart[lane] = mp;
        __syncthreads();
        float mrow = fmaxf(mpart[l15], mpart[16 + l15]);
        if (half == 0) mLDS[l15] = mrow;

        // e = exp(fv - m), row-major in LDS
#pragma unroll
        for (int q = 0; q < 24; ++q) {
            int j = half * 24 + q;
            eLDS[l15 * FVS + j] = __expf(fvLDS[l15 * FVS + j] - mrow);
        }
        __syncthreads();

        // A fragments: 16x4 f32 layout (M = l15; K = 4c + v + 2*half)
        v2f A[12];
#pragma unroll
        for (int c = 0; c < 12; ++c) {
            int k0 = 4 * c + 2 * half;
            A[c].x = eLDS[l15 * FVS + k0];
            A[c].y = eLDS[l15 * FVS + k0 + 1];
        }

        // s = e @ P^T  — 36 x v_wmma_f32_16x16x4_f32
        v8f acc[3] = {};
#pragma unroll
        for (int jt = 0; jt < 3; ++jt)
#pragma unroll
            for (int c = 0; c < 12; ++c)
                acc[jt] = __builtin_amdgcn_wmma_f32_16x16x4_f32(
                    false, A[c], false, Bf[c][jt], (short)0, acc[jt], false, false);

        __syncthreads();   // all reads of old fvLDS done before overwrite

        // fv' = log(s) + m + emit  (C layout: M = v + 8*half, N = l15)
#pragma unroll
        for (int jt = 0; jt < 3; ++jt)
#pragma unroll
            for (int v = 0; v < 8; ++v) {
                int M = v + 8 * half;
                float val = __logf(acc[jt][v]) + mLDS[M] + emit[jt][v];
                fvLDS[M * FVS + jt * 16 + l15] = val;
            }
        __syncthreads();

        // Capture terminal forward score when this batch's sequence ends here.
        if (lane < 16) {
            int b = bb + lane;
            if (seq_len[b] == t + 1) {
                float mm = -3.402823466e38f;
                for (int j = 0; j < KTAG; ++j)
                    mm = fmaxf(mm, fvLDS[lane * FVS + j] + tS[j]);
                float ss = 0.0f;
                for (int j = 0; j < KTAG; ++j)
                    ss += __expf(fvLDS[lane * FVS + j] + tS[j] - mm);
                fwd[b] = mm + __logf(ss);
            }
        }
    }
}

// ---------------------------------------------------------------------------
// Kernel 2: gold path score, one wave per batch
// ---------------------------------------------------------------------------
__global__ __launch_bounds__(32) void crf_gold(const float* __restrict__ feats,
                                               const int*   __restrict__ tags,
                                               const int*   __restrict__ seq_len,
                                               const float* __restrict__ trans,
                                               float*       __restrict__ gold,
                                               int T) {
    int b = blockIdx.x, lane = threadIdx.x;
    int L = seq_len[b];
    float s = 0.0f;
    for (int t = lane; t < L; t += 32) {
        int cur  = tags[b * T + t];
        int prev = (t == 0) ? START : tags[b * T + t - 1];
        s += trans[cur * KTAG + prev] +
             feats[(size_t)b * T * KTAG + (size_t)t * KTAG + cur];
    }
#pragma unroll
    for (int off = 16; off > 0; off >>= 1) s += __shfl_down(s, off, 32);
    if (lane == 0) {
        int last = tags[b * T + L - 1];                   // L >= 1
        gold[b] = s + trans[STOP * KTAG + last];
    }
}

// ---------------------------------------------------------------------------
// Kernel 3: mean(forward - gold)
// ---------------------------------------------------------------------------
__global__ void crf_reduce(const float* __restrict__ fwd, const float* __restrict__ gold,
                           float* __restrict__ out, int B) {
    __shared__ float buf[256];
    int tid = threadIdx.x;
    float s = 0.0f;
    for (int i = tid; i < B; i += 256) s += fwd[i] - gold[i];
    buf[tid] = s;
    __syncthreads();
    for (int off = 128; off > 0; off >>= 1) {
        if (tid < off) buf[tid] += buf[tid + off];
        __syncthreads();
    }
    if (tid == 0) out[0] = buf[0] / (float)B;
}

// ---------------------------------------------------------------------------
extern "C" void kernel_launch(void* const* d_in, const int* in_sizes, int n_in,
                              void* d_out, int out_size, void* d_ws, size_t ws_size,
                              hipStream_t stream) {
    const float* feats   = (const float*)d_in[0];   // (B,T,48)
    const int*   tags    = (const int*)  d_in[1];   // (B,T)
    const int*   seq_len = (const int*)  d_in[2];   // (B,)
    const float* trans   = (const float*)d_in[3];   // (48,48)

    const int B = in_sizes[2];
    const int T = in_sizes[1] / B;

    float* Pt   = (float*)d_ws;                 // 48*48
    float* fwd  = Pt + KTAG * KTAG;             // B
    float* gold = fwd + B;                      // B

    crf_prep  <<<(KTAG * KTAG + 255) / 256, 256, 0, stream>>>(trans, Pt);
    crf_scan  <<<B / 16, 32, 0, stream>>>(feats, seq_len, trans, Pt, fwd, T);
    crf_gold  <<<B, 32, 0, stream>>>(feats, tags, seq_len, trans, gold, T);
    crf_reduce<<<1, 256, 0, stream>>>(fwd, gold, (float*)d_out, B);
}